// CustomEmbeddingCollection_58291296141452
// MI455X (gfx1250) — compile-verified
//
#include <hip/hip_runtime.h>

// Embedding-cache gather for MI455X (gfx1250, wave32).
// out[n, 0:64] = hit ? weight_gpu[pos[n]] : weight_cpu[indices[n]]
// pos = inverse_map[indices[n]], inverse_map built from gpu_cache_rows.
//
// Main gather uses the CDNA5 async global<->LDS path (ASYNCcnt):
//   global_load_async_to_lds_b128  ->  s_wait_asynccnt 0  ->
//   global_store_async_from_lds_b128 th:TH_STORE_NT
// Row data never passes through VGPRs.

typedef __attribute__((ext_vector_type(4))) float v4f;
typedef __attribute__((ext_vector_type(4))) int   v4i;

#define EMB_DIM          64
#define ROW_BYTES        256  // 64 floats
#define VEC_PER_ROW      16   // 16 x 16B chunks per row
#define ROWS_PER_BLOCK   16
#define BLOCK            256  // 8 waves (wave32); one half-wave per row

// ---------------------------------------------------------------------------
// Kernel 1: fill inverse map with -1 (wide b128 stores).
// ---------------------------------------------------------------------------
__global__ void fill_inv_kernel(int* __restrict__ inv, int num_emb) {
    int i4   = blockIdx.x * blockDim.x + threadIdx.x;
    int base = i4 << 2;
    if (base + 3 < num_emb) {
        v4i m1 = {-1, -1, -1, -1};
        *(v4i*)(inv + base) = m1;
    } else {
        for (int k = base; k < num_emb; ++k) inv[k] = -1;
    }
}

// ---------------------------------------------------------------------------
// Kernel 2: scatter cache slot ids: inv[gpu_cache_rows[i]] = i.
// Rows are unique (permutation subset) -> plain stores, no atomics.
// ---------------------------------------------------------------------------
__global__ void scatter_inv_kernel(const int* __restrict__ cache_rows,
                                   int* __restrict__ inv, int n_cache) {
    int i = blockIdx.x * blockDim.x + threadIdx.x;
    if (i < n_cache) inv[cache_rows[i]] = i;
}

// ---------------------------------------------------------------------------
// Kernel 3: main gather via async global->LDS->global DMA path.
// 256-thread block handles 16 rows; 16 lanes per row, 16B per lane.
// Each thread owns a private 16B LDS staging slot, so only a per-wave
// s_wait_asynccnt is needed between the async load and the async store.
// ---------------------------------------------------------------------------
__global__ void gather_async_kernel(const int*   __restrict__ indices,
                                    const float* __restrict__ wcpu,
                                    const float* __restrict__ wgpu,
                                    const int*   __restrict__ inv,
                                    float*       __restrict__ out,
                                    int n) {
    __shared__ int s_idx[ROWS_PER_BLOCK];
    __shared__ int s_pos[ROWS_PER_BLOCK];
    __shared__ v4f stage[BLOCK];          // 4KB: 16B private slot per thread

    const int rbase = blockIdx.x * ROWS_PER_BLOCK;

    if (threadIdx.x < ROWS_PER_BLOCK) {
        int r   = rbase + threadIdx.x;
        int idx = 0, pos = -1;
        if (r < n) {
            idx = __builtin_nontemporal_load(&indices[r]); // read-once stream
            pos = inv[idx];                                // L2-resident map
        }
        s_idx[threadIdx.x] = idx;
        s_pos[threadIdx.x] = pos;
    }
    __syncthreads();

    const int local = threadIdx.x >> 4;   // row within block: 0..15
    const int sub   = threadIdx.x & 15;   // 16B chunk within row: 0..15
    const int row   = rbase + local;
    if (row >= n) return;

    const int idx = s_idx[local];
    const int pos = s_pos[local];

    const char* src = (pos >= 0)
        ? (const char*)(wgpu + (size_t)pos * EMB_DIM)
        : (const char*)(wcpu + (size_t)idx * EMB_DIM);
    src += sub * 16;

    // LDS byte address = low 32 bits of the generic pointer (ISA 10.2:
    // LDS aperture keeps the LDS offset in addr[31:0]).
    unsigned lds_slot = (unsigned)(size_t)&stage[threadIdx.x];

    // Memory -> LDS, no VGPR data path. Tracked with ASYNCcnt.
    asm volatile("global_load_async_to_lds_b128 %0, %1, off"
                 :: "v"(lds_slot), "v"(src)
                 : "memory");

    // Wait for this wave's async LDS writes to land.
    asm volatile("s_wait_asynccnt 0x0" ::: "memory");

    char* dst = (char*)out + (size_t)row * ROW_BYTES + sub * 16;

    // LDS -> memory, non-temporal: keep streamed output out of L2.
    asm volatile("global_store_async_from_lds_b128 %0, %1, off th:TH_STORE_NT"
                 :: "v"(dst), "v"(lds_slot)
                 : "memory");
    // s_endpgm performs an implicit wait-idle, draining remaining ASYNCcnt.
}

// ---------------------------------------------------------------------------
// Fallback (ws too small for the inverse map): weight_gpu is an exact copy of
// weight_cpu[gpu_cache_rows], so out == weight_cpu[indices] bit-for-bit.
// ---------------------------------------------------------------------------
__global__ void gather_direct_kernel(const int*   __restrict__ indices,
                                     const float* __restrict__ wcpu,
                                     float*       __restrict__ out,
                                     int n) {
    __shared__ int s_idx[ROWS_PER_BLOCK];
    const int rbase = blockIdx.x * ROWS_PER_BLOCK;

    if (threadIdx.x < ROWS_PER_BLOCK) {
        int r = rbase + threadIdx.x;
        s_idx[threadIdx.x] = (r < n) ? __builtin_nontemporal_load(&indices[r]) : 0;
    }
    __syncthreads();

    const int local = threadIdx.x >> 4;
    const int sub   = threadIdx.x & 15;
    const int row   = rbase + local;
    if (row >= n) return;

    const int idx = s_idx[local];
    v4f v = ((const v4f*)(wcpu + (size_t)idx * EMB_DIM))[sub];
    __builtin_nontemporal_store(v, (v4f*)out + (size_t)row * VEC_PER_ROW + sub);
}

// ---------------------------------------------------------------------------
extern "C" void kernel_launch(void* const* d_in, const int* in_sizes, int n_in,
                              void* d_out, int out_size, void* d_ws, size_t ws_size,
                              hipStream_t stream) {
    const int*   indices    = (const int*)d_in[0];
    const float* weight_cpu = (const float*)d_in[1];
    const float* weight_gpu = (const float*)d_in[2];
    const int*   cache_rows = (const int*)d_in[3];

    const int n_lookups = in_sizes[0];
    const int num_emb   = in_sizes[1] / EMB_DIM;
    const int n_cache   = in_sizes[3];
    float* out = (float*)d_out;

    const int row_blocks = (n_lookups + ROWS_PER_BLOCK - 1) / ROWS_PER_BLOCK;

    if (ws_size >= (size_t)num_emb * sizeof(int)) {
        int* inv = (int*)d_ws;

        // 1) inv[:] = -1  (each thread fills 4 ints with one b128 store)
        int fill_items  = (num_emb + 3) >> 2;
        int fill_blocks = (fill_items + BLOCK - 1) / BLOCK;
        fill_inv_kernel<<<fill_blocks, BLOCK, 0, stream>>>(inv, num_emb);

        // 2) inv[cache_rows[i]] = i
        int sc_blocks = (n_cache + BLOCK - 1) / BLOCK;
        scatter_inv_kernel<<<sc_blocks, BLOCK, 0, stream>>>(cache_rows, inv, n_cache);

        // 3) gather with hit/miss select via async global<->LDS DMA
        gather_async_kernel<<<row_blocks, BLOCK, 0, stream>>>(
            indices, weight_cpu, weight_gpu, inv, out, n_lookups);
    } else {
        gather_direct_kernel<<<row_blocks, BLOCK, 0, stream>>>(
            indices, weight_cpu, out, n_lookups);
    }
}